// cvx_knapsack_solver_61658550502067
// MI455X (gfx1250) — compile-verified
//
#include <hip/hip_runtime.h>
#include <cstdint>

// ---------------- problem constants (match reference) ----------------
static constexpr int   kNItems       = 64;
static constexpr int   kLanesPerRow  = 4;
static constexpr int   kItemsPerLane = kNItems / kLanesPerRow;  // 16
static constexpr int   kBisect       = 60;
static constexpr int   kPgd          = 120;
static constexpr float kCap          = 102.0f;   // 0.3 * sum(weights)
static constexpr float kEta          = 0.1f;

// ---------------- wave32 quad (4-lane) butterflies via ds_swizzle ----------------
template <int IMM>
__device__ __forceinline__ float qswz(float v) {
  return __int_as_float(__builtin_amdgcn_ds_swizzle(__float_as_int(v), IMM));
}
// full butterfly within 4 consecutive lanes -> result broadcast to all 4
__device__ __forceinline__ float qsum(float v) {
  v += qswz<0x80B1>(v);   // quad_perm(1,0,3,2)  == xor 1
  v += qswz<0x804E>(v);   // quad_perm(2,3,0,1)  == xor 2
  return v;
}
__device__ __forceinline__ float qmax(float v) {
  v = fmaxf(v, qswz<0x80B1>(v));
  v = fmaxf(v, qswz<0x804E>(v));
  return v;
}

// g(lam) = sum_j w_j * clip(y_j - lam*w_j, 0, 1) over the full row (quad-reduced)
__device__ __forceinline__ float eval_g(const float y[kItemsPerLane],
                                        const float w[kItemsPerLane],
                                        float lam) {
  float a0 = 0.f, a1 = 0.f, a2 = 0.f, a3 = 0.f;
#pragma unroll
  for (int k = 0; k < kItemsPerLane; k += 4) {
    float t0 = __builtin_amdgcn_fmed3f(__builtin_fmaf(-lam, w[k + 0], y[k + 0]), 0.f, 1.f);
    float t1 = __builtin_amdgcn_fmed3f(__builtin_fmaf(-lam, w[k + 1], y[k + 1]), 0.f, 1.f);
    float t2 = __builtin_amdgcn_fmed3f(__builtin_fmaf(-lam, w[k + 2], y[k + 2]), 0.f, 1.f);
    float t3 = __builtin_amdgcn_fmed3f(__builtin_fmaf(-lam, w[k + 3], y[k + 3]), 0.f, 1.f);
    a0 = __builtin_fmaf(t0, w[k + 0], a0);
    a1 = __builtin_fmaf(t1, w[k + 1], a1);
    a2 = __builtin_fmaf(t2, w[k + 2], a2);
    a3 = __builtin_fmaf(t3, w[k + 3], a3);
  }
  return qsum((a0 + a1) + (a2 + a3));
}

// Euclidean projection of the row held in xy[] onto {0<=x<=1, w@x<=kCap}.
// In/out: xy[] (enters as y, leaves as x = clip(y - lam*w, 0, 1)).
__device__ __forceinline__ void project_row(float xy[kItemsPerLane],
                                            const float w[kItemsPerLane],
                                            const float rw[kItemsPerLane]) {
  // upper bracket: max over row of relu(y)/w, + 1e-3
  float m = 0.f;
#pragma unroll
  for (int k = 0; k < kItemsPerLane; ++k)
    m = fmaxf(m, fmaxf(xy[k], 0.f) * rw[k]);
  float hi = qmax(m) + 1e-3f;

  const bool feas = (eval_g(xy, w, 0.f) <= kCap);  // constraint inactive -> lam = 0

  float lo = 0.f;
#pragma unroll 1
  for (int it = 0; it < kBisect; ++it) {
    const float mid  = 0.5f * (lo + hi);
    const bool  over = (eval_g(xy, w, mid) > kCap);
    lo = over ? mid : lo;
    hi = over ? hi : mid;
  }
  float lam = 0.5f * (lo + hi);
  lam = feas ? 0.f : lam;

#pragma unroll
  for (int k = 0; k < kItemsPerLane; ++k)
    xy[k] = __builtin_amdgcn_fmed3f(__builtin_fmaf(-lam, w[k], xy[k]), 0.f, 1.f);
}

// vector-of-4-int pointer types matching the async-to-LDS builtin's signature
typedef int v4i __attribute__((vector_size(16)));
typedef __attribute__((address_space(1))) v4i gv4i;   // global
typedef __attribute__((address_space(3))) v4i lv4i;   // LDS

__global__ __launch_bounds__(256) void knap_pgd_kernel(
    const float* __restrict__ costs, const float* __restrict__ weights,
    float* __restrict__ out, int n_rows) {
  __shared__ float wlds[kNItems];

  // ---- stage the broadcast weight vector into LDS (CDNA5 async path) ----
#if __has_builtin(__builtin_amdgcn_global_load_async_to_lds_b128)
  {
    const int lt = (int)(threadIdx.x & 31u);
    if (lt < (kNItems / 4)) {   // 16 lanes x 16B = 64 floats
      gv4i* gp = (gv4i*)(uintptr_t)(weights + lt * 4);
      lv4i* lp = (lv4i*)(uint32_t)(uintptr_t)(&wlds[lt * 4]);
      __builtin_amdgcn_global_load_async_to_lds_b128(gp, lp, 0, 0);
    }
#if __has_builtin(__builtin_amdgcn_s_wait_asynccnt)
    __builtin_amdgcn_s_wait_asynccnt(0);
#else
    asm volatile("s_wait_asynccnt 0" ::: "memory");
#endif
  }
#else
  if (threadIdx.x < (unsigned)kNItems) wlds[threadIdx.x] = weights[threadIdx.x];
#endif
  __syncthreads();

  const int q   = (int)(threadIdx.x & (kLanesPerRow - 1));       // quad lane
  const int row = (int)(blockIdx.x * (blockDim.x / kLanesPerRow) + (threadIdx.x >> 2));
  if (row >= n_rows) return;   // whole quads retire together

  float c[kItemsPerLane], w[kItemsPerLane], rw[kItemsPerLane], xy[kItemsPerLane];

  // coalesced b128 loads: each lane owns 16 contiguous items of its row
  const float4* cp = reinterpret_cast<const float4*>(costs + (size_t)row * kNItems +
                                                     q * kItemsPerLane);
  const float4* wp = reinterpret_cast<const float4*>(&wlds[q * kItemsPerLane]);
#pragma unroll
  for (int i = 0; i < 4; ++i) {
    float4 cv = cp[i];
    c[4 * i + 0] = cv.x; c[4 * i + 1] = cv.y; c[4 * i + 2] = cv.z; c[4 * i + 3] = cv.w;
    float4 wv = wp[i];
    w[4 * i + 0] = wv.x; w[4 * i + 1] = wv.y; w[4 * i + 2] = wv.z; w[4 * i + 3] = wv.w;
  }
#pragma unroll
  for (int k = 0; k < kItemsPerLane; ++k) rw[k] = 1.0f / w[k];

  // x0 = project(clip(costs, 0, 1))
#pragma unroll
  for (int k = 0; k < kItemsPerLane; ++k)
    xy[k] = __builtin_amdgcn_fmed3f(c[k], 0.f, 1.f);
  project_row(xy, w, rw);

  // PGD: x <- project(x + eta * (c - x / ||x||))
#pragma unroll 1
  for (int it = 0; it < kPgd; ++it) {
    float s0 = 0.f, s1 = 0.f, s2 = 0.f, s3 = 0.f;
#pragma unroll
    for (int k = 0; k < kItemsPerLane; k += 4) {
      s0 = __builtin_fmaf(xy[k + 0], xy[k + 0], s0);
      s1 = __builtin_fmaf(xy[k + 1], xy[k + 1], s1);
      s2 = __builtin_fmaf(xy[k + 2], xy[k + 2], s2);
      s3 = __builtin_fmaf(xy[k + 3], xy[k + 3], s3);
    }
    const float nrm = sqrtf(qsum((s0 + s1) + (s2 + s3)) + 1e-12f);
    const float inv = 1.0f / nrm;   // MU == 1
#pragma unroll
    for (int k = 0; k < kItemsPerLane; ++k) {
      const float grad = __builtin_fmaf(-inv, xy[k], c[k]);  // c - x/||x||
      xy[k] = __builtin_fmaf(kEta, grad, xy[k]);             // y = x + eta*grad
    }
    project_row(xy, w, rw);
  }

  float4* op = reinterpret_cast<float4*>(out + (size_t)row * kNItems + q * kItemsPerLane);
#pragma unroll
  for (int i = 0; i < 4; ++i) {
    float4 v;
    v.x = xy[4 * i + 0]; v.y = xy[4 * i + 1]; v.z = xy[4 * i + 2]; v.w = xy[4 * i + 3];
    op[i] = v;
  }
}

extern "C" void kernel_launch(void* const* d_in, const int* in_sizes, int n_in,
                              void* d_out, int out_size, void* d_ws, size_t ws_size,
                              hipStream_t stream) {
  const float* costs   = (const float*)d_in[0];
  const float* weights = (const float*)d_in[1];
  float* out = (float*)d_out;
  (void)n_in; (void)out_size; (void)d_ws; (void)ws_size;

  const int n_rows = in_sizes[0] / kNItems;               // 16384
  const int block  = 256;                                 // 8 wave32s
  const int rows_per_block = block / kLanesPerRow;        // 64
  const int grid = (n_rows + rows_per_block - 1) / rows_per_block;
  knap_pgd_kernel<<<grid, block, 0, stream>>>(costs, weights, out, n_rows);
}